// MotionCorrect_27736898798200
// MI455X (gfx1250) — compile-verified
//
#include <hip/hip_runtime.h>
#include <math.h>
#include <stdint.h>

// ---------------- problem constants ----------------
#define H0f 3072            // full frame H=W
#define HCc 2048            // crop H=W
#define MS  10              // max shift
#define SCX 512             // crop offset (H0-HC)/2

typedef float v2f __attribute__((ext_vector_type(2)));
typedef float v8f __attribute__((ext_vector_type(8)));

// ---------------- workspace layout (bytes) ----------------
#define SCAL_OFF   0                      // 8 doubles: tmean, fmean, tvar
#define RPART_OFF  64                     // 1024*3 doubles
#define BAND_OFF   (RPART_OFF + 24576)    // 2*21*2048 doubles
#define DENOM_OFF  (BAND_OFF + 688128)    // 441 doubles
#define CCPART_OFF (DENOM_OFF + 4096)     // 1024*1024 floats
#define CC_OFF     (CCPART_OFF + 4194304) // 1024 floats (32x32 padded cc)
#define SH_OFF     (CC_OFF + 4096)        // 2 floats

// ---------------- CDNA5 async global->LDS helpers ----------------
__device__ __forceinline__ uint32_t lds_off32(const void* p) {
    // generic shared-aperture pointer: low 32 bits are the LDS byte address
    return (uint32_t)(uintptr_t)p;
}
__device__ __forceinline__ void async_ld_b128(uint32_t lds_addr, const float* gaddr) {
    asm volatile("global_load_async_to_lds_b128 %0, %1, off"
                 :: "v"(lds_addr), "v"(gaddr) : "memory");
}
__device__ __forceinline__ void async_ld_b32(uint32_t lds_addr, const float* gaddr) {
    asm volatile("global_load_async_to_lds_b32 %0, %1, off"
                 :: "v"(lds_addr), "v"(gaddr) : "memory");
}
__device__ __forceinline__ void wait_async0() {
    asm volatile("s_wait_asynccnt 0x0" ::: "memory");
}

// ================= global stats (tmpl mean/var, fr mean) =================
__global__ void k_reduce1(const float* __restrict__ fr, const float* __restrict__ tp,
                          double* __restrict__ rpart) {
    __shared__ double sd[768];
    int tid = threadIdx.x;
    double st = 0.0, st2 = 0.0, sf = 0.0;
    for (long idx = (long)blockIdx.x * blockDim.x + tid; idx < (long)HCc * HCc;
         idx += (long)gridDim.x * blockDim.x) {
        int i = (int)(idx >> 11), j = (int)(idx & 2047);
        long off = (long)(SCX + i) * H0f + SCX + j;
        double t = tp[off], f = fr[off];
        st += t; st2 += t * t; sf += f;
    }
    sd[tid] = st; sd[256 + tid] = st2; sd[512 + tid] = sf;
    __syncthreads();
    for (int s = 128; s > 0; s >>= 1) {
        if (tid < s) {
            sd[tid] += sd[tid + s];
            sd[256 + tid] += sd[256 + tid + s];
            sd[512 + tid] += sd[512 + tid + s];
        }
        __syncthreads();
    }
    if (tid == 0) {
        rpart[blockIdx.x * 3 + 0] = sd[0];
        rpart[blockIdx.x * 3 + 1] = sd[256];
        rpart[blockIdx.x * 3 + 2] = sd[512];
    }
}

__global__ void k_reduce2(const double* __restrict__ rpart, double* __restrict__ scal) {
    __shared__ double sd[768];
    int tid = threadIdx.x;
    double a = 0.0, b = 0.0, c = 0.0;
    for (int k = tid; k < 1024; k += 256) {
        a += rpart[k * 3 + 0]; b += rpart[k * 3 + 1]; c += rpart[k * 3 + 2];
    }
    sd[tid] = a; sd[256 + tid] = b; sd[512 + tid] = c;
    __syncthreads();
    for (int s = 128; s > 0; s >>= 1) {
        if (tid < s) {
            sd[tid] += sd[tid + s];
            sd[256 + tid] += sd[256 + tid + s];
            sd[512 + tid] += sd[512 + tid + s];
        }
        __syncthreads();
    }
    if (tid == 0) {
        const double N = (double)HCc * (double)HCc;
        double tm = sd[0] / N;
        double fm = sd[512] / N;
        scal[0] = tm;
        scal[1] = fm;
        scal[2] = sd[256] - N * tm * tm + 1e-8;   // t_var = sum t^2 - N*tm^2 + eps
    }
}

// ============ sliding column-band sums of fr and fr^2 (f64, like ref cumsum) ============
__global__ void k_colband(const float* __restrict__ fr, double* __restrict__ band) {
    int j = blockIdx.x * blockDim.x + threadIdx.x;
    if (j >= HCc) return;
    const int KH = HCc - 2 * MS;   // 2028
    double s1 = 0.0, s2 = 0.0;
    for (int i = 0; i < KH; ++i) {
        double v = fr[(long)(SCX + i) * H0f + SCX + j];
        s1 += v; s2 += v * v;
    }
    for (int a = 0; a <= 2 * MS; ++a) {
        band[(long)a * HCc + j] = s1;
        band[21L * HCc + (long)a * HCc + j] = s2;
        if (a < 2 * MS) {
            double vn = fr[(long)(SCX + a + KH) * H0f + SCX + j];
            double vo = fr[(long)(SCX + a) * H0f + SCX + j];
            s1 += vn - vo; s2 += vn * vn - vo * vo;
        }
    }
}

// ============ 21x21 window variance -> denom = sqrt(t_var * imgs_var) ============
__global__ void k_winvar(const double* __restrict__ band, const double* __restrict__ scal,
                         double* __restrict__ denom) {
    __shared__ double sd[512];
    int a = blockIdx.x / 21, b = blockIdx.x % 21;
    int tid = threadIdx.x;
    const int KW = HCc - 2 * MS;   // 2028
    double s1 = 0.0, s2 = 0.0;
    for (int j = tid; j < KW; j += 256) {
        s1 += band[(long)a * HCc + b + j];
        s2 += band[21L * HCc + (long)a * HCc + b + j];
    }
    sd[tid] = s1; sd[256 + tid] = s2;
    __syncthreads();
    for (int s = 128; s > 0; s >>= 1) {
        if (tid < s) { sd[tid] += sd[tid + s]; sd[256 + tid] += sd[256 + tid + s]; }
        __syncthreads();
    }
    if (tid == 0) {
        double kk = (double)KW * (double)KW;
        double lm1 = sd[0] / kk, lm2 = sd[256] / kk;
        double var = lm2 - lm1 * lm1 / ((double)HCc * (double)HCc) + 1e-8;
        if (var < 0.0) var = 0.0;
        denom[blockIdx.x] = sqrt(scal[2] * var);
    }
}

// ================= windowed circular correlation via WMMA f32 16x16x4 =================
// cc[u+10][v+10] = sum_{i,c} tpl[(i-u)%2048, c] * fr[i, (c+v)%2048]  (raw values)
// Per row i this is a 32x32-padded GEMM with K over columns.
#define IB   16     // rows per workgroup
#define CWG  256    // columns per workgroup (4 waves x 64)
#define CWV  64     // columns per wave
#define TPS  260    // LDS pitch: 260%64==4 -> conflict-free A loads; rows 16B-aligned for b128
#define FRS  277
#define ZIDX (36 * TPS + IB * FRS)   // zero-pad region (reads for padded m/n >= 21)

__global__ __launch_bounds__(128) void k_corr(const float* __restrict__ fr,
                                              const float* __restrict__ tp,
                                              float* __restrict__ ccpart) {
    __shared__ float smem[36 * TPS + IB * FRS + 32];   // ~55.4 KB
    const int tid  = threadIdx.x;
    const int lane = tid & 31;
    const int wv   = tid >> 5;
    const int cbase = blockIdx.x * CWG;
    const int i0    = blockIdx.y * IB;

    // zero pad region (selected by lanes whose padded tile row/col is invalid)
    if (tid < 32) smem[ZIDX + tid] = 0.f;

    // --- async stage: template rows (i0-10 .. i0+25) wrapped, 256 cols, b128 ---
    for (int e = tid; e < 36 * CWG / 4; e += 128) {
        int fl = e * 4;
        int rr = fl >> 8;          // /256
        int cc = fl & 255;
        int gi = (i0 - MS + rr) & (HCc - 1);
        async_ld_b128(lds_off32(&smem[rr * TPS + cc]),
                      &tp[(long)(SCX + gi) * H0f + SCX + cbase + cc]);
    }
    // --- async stage: fr rows i0..i0+15, cols cbase-10 .. cbase+265 wrapped, b32 ---
    for (int e = tid; e < IB * 276; e += 128) {
        int rr = e / 276;
        int cc = e % 276;
        int gj = (cbase - MS + cc) & (HCc - 1);
        async_ld_b32(lds_off32(&smem[36 * TPS + rr * FRS + cc]),
                     &fr[(long)(SCX + i0 + rr) * H0f + SCX + gj]);
    }
    wait_async0();
    __syncthreads();

    v8f c00 = {}, c01 = {}, c10 = {}, c11 = {};
    const int M0 = lane & 15;                // A-tile row / B-tile col within 16
    const int kk = (lane < 16) ? 0 : 2;      // K pair held by this lane
    const int cw = wv * CWV;
    const bool hi_ok = (M0 <= 4);            // padded tiles valid only for m,n = 16..20

    for (int ii = 0; ii < IB; ++ii) {
        const int frbase = 36 * TPS + ii * FRS;
        #pragma unroll 4
        for (int ks = 0; ks < CWV / 4; ++ks) {
            const int cbl = cw + ks * 4 + kk;
            // A tile 0: m = M0, row rr = ii + 20 - m
            const int a0i = (ii + 20 - M0) * TPS + cbl;
            v2f a0; a0.x = smem[a0i]; a0.y = smem[a0i + 1];
            // A tile 1: m = 16 + M0 (zero-pad read when m > 20): branchless addr select
            const int a1i = hi_ok ? ((ii + 4 - M0) * TPS + cbl) : ZIDX;
            v2f a1; a1.x = smem[a1i]; a1.y = smem[a1i + 1];
            // B tile 0: n = M0; LDS col = cw + ks*4 + k + n (halo folded in)
            const int b0i = frbase + cbl + M0;
            v2f b0; b0.x = smem[b0i]; b0.y = smem[b0i + 1];
            // B tile 1: n = 16 + M0 (zero-pad read when n > 20)
            const int b1i = hi_ok ? (b0i + 16) : ZIDX;
            v2f b1; b1.x = smem[b1i]; b1.y = smem[b1i + 1];

            c00 = __builtin_amdgcn_wmma_f32_16x16x4_f32(false, a0, false, b0, (short)0, c00, false, false);
            c01 = __builtin_amdgcn_wmma_f32_16x16x4_f32(false, a0, false, b1, (short)0, c01, false, false);
            c10 = __builtin_amdgcn_wmma_f32_16x16x4_f32(false, a1, false, b0, (short)0, c10, false, false);
            c11 = __builtin_amdgcn_wmma_f32_16x16x4_f32(false, a1, false, b1, (short)0, c11, false, false);
        }
    }
    __syncthreads();

    // per-wave 32x32 tiles -> LDS (reuse staging), reduce 4 waves -> WG partial
    float* red = smem;
    const int mrow = (lane >> 4) * 8;   // C/D layout: vgpr r, lanes16-31 are M+8
    const int n0   = lane & 15;
    const int base = wv * 1024;
    #pragma unroll
    for (int r = 0; r < 8; ++r) {
        int m = mrow + r;
        red[base + m * 32 + n0]             = c00[r];
        red[base + m * 32 + 16 + n0]        = c01[r];
        red[base + (m + 16) * 32 + n0]      = c10[r];
        red[base + (m + 16) * 32 + 16 + n0] = c11[r];
    }
    __syncthreads();
    const long wg = (long)blockIdx.y * gridDim.x + blockIdx.x;
    for (int e = tid; e < 1024; e += 128) {
        float s = red[e] + red[1024 + e] + red[2048 + e] + red[3072 + e];
        ccpart[wg * 1024 + e] = s;
    }
}

// ============ deterministic partial reduction + mean fixup + abs ============
__global__ void k_ccred(const float* __restrict__ ccpart, const double* __restrict__ scal,
                        float* __restrict__ cc) {
    int e = blockIdx.x * blockDim.x + threadIdx.x;
    if (e >= 1024) return;
    float s = 0.f;
    for (int wg = 0; wg < 1024; ++wg) s += ccpart[(long)wg * 1024 + e];
    double fix = (double)HCc * (double)HCc * scal[0] * scal[1];  // N*tmean*fmean
    cc[e] = fabsf((float)((double)s - fix));                     // ref takes |ifft|
}

// ============ ncc, argmax, fractional log-parabola peak ============
__global__ void k_peak(const float* __restrict__ cc, const double* __restrict__ denom,
                       float* __restrict__ sh, float* __restrict__ out_tail) {
    __shared__ float nccS[441];
    int tid = threadIdx.x;
    if (tid < 441) {
        int a = tid / 21, b = tid % 21;
        float v = (float)((double)cc[a * 32 + b] / denom[tid]);
        if (v != v) v = 0.f;   // NaN -> 0 per reference
        nccS[tid] = v;
    }
    __syncthreads();
    if (tid == 0) {
        int am = 0;
        float mx = nccS[0];
        for (int k = 1; k < 441; ++k) {
            if (nccS[k] > mx) { mx = nccS[k]; am = k; }
        }
        int sx = am / 21, sy = am % 21;
        auto L = [&](int x, int y) {
            x = min(max(x, 0), 20); y = min(max(y, 0), 20);
            return logf(nccS[x * 21 + y]);
        };
        float l0   = L(sx, sy);
        float lxm1 = L(sx - 1, sy), lxp1 = L(sx + 1, sy);
        float lym1 = L(sx, sy - 1), lyp1 = L(sx, sy + 1);
        float shx = -(float)(sx - MS) - (lxm1 - lxp1) / (2.f * lxm1 - 4.f * l0 + 2.f * lxp1);
        float shy = -(float)(sy - MS) - (lym1 - lyp1) / (2.f * lym1 - 4.f * l0 + 2.f * lyp1);
        sh[0] = shx; sh[1] = shy;
        out_tail[0] = shx; out_tail[1] = shy;
    }
}

// ============ fused bilinear translate + transposed (column-major) flatten ============
__device__ __forceinline__ float samp(const float* __restrict__ fr, int ri, int ci) {
    bool valid = (ri >= 0) && (ri < H0f) && (ci >= 0) && (ci < H0f);
    int rc = min(max(ri, 0), H0f - 1);
    int cc = min(max(ci, 0), H0f - 1);
    float v = fr[(long)rc * H0f + cc];
    return valid ? v : 0.f;
}

__global__ __launch_bounds__(256) void k_trans(const float* __restrict__ fr,
                                               const float* __restrict__ sh,
                                               float* __restrict__ out) {
    __shared__ float tile[32][33];
    const float dx = sh[0], dy = sh[1];
    const int tx = threadIdx.x, ty = threadIdx.y;       // block (32, 8)
    const int rb = blockIdx.y * 32, cb = blockIdx.x * 32;
    #pragma unroll
    for (int q = 0; q < 4; ++q) {
        int r = rb + ty + q * 8;
        int c = cb + tx;
        float rr = (float)r - dy;
        float cc = (float)c - dx;
        float r0f = floorf(rr), c0f = floorf(cc);
        int r0 = (int)r0f, c0 = (int)c0f;
        float wr = rr - r0f, wc = cc - c0f;
        float v00 = samp(fr, r0, c0),     v01 = samp(fr, r0, c0 + 1);
        float v10 = samp(fr, r0 + 1, c0), v11 = samp(fr, r0 + 1, c0 + 1);
        tile[ty + q * 8][tx] = v00 * (1.f - wr) * (1.f - wc) + v01 * (1.f - wr) * wc
                             + v10 * wr * (1.f - wc)         + v11 * wr * wc;
    }
    __syncthreads();
    #pragma unroll
    for (int q = 0; q < 4; ++q) {
        int rw = rb + tx;
        int cw = cb + ty + q * 8;
        out[(long)cw * H0f + rw] = tile[tx][ty + q * 8];   // out = reshape(fr_corr.T, -1)
    }
}

// ================================ launcher ================================
extern "C" void kernel_launch(void* const* d_in, const int* in_sizes, int n_in,
                              void* d_out, int out_size, void* d_ws, size_t ws_size,
                              hipStream_t stream) {
    const float* fr = (const float*)d_in[0];
    const float* tp = (const float*)d_in[1];
    float* out = (float*)d_out;
    char* ws = (char*)d_ws;

    double* scal   = (double*)(ws + SCAL_OFF);
    double* rpart  = (double*)(ws + RPART_OFF);
    double* band   = (double*)(ws + BAND_OFF);
    double* denom  = (double*)(ws + DENOM_OFF);
    float*  ccpart = (float*)(ws + CCPART_OFF);
    float*  cc     = (float*)(ws + CC_OFF);
    float*  sh     = (float*)(ws + SH_OFF);

    (void)in_sizes; (void)n_in; (void)out_size; (void)ws_size;

    k_reduce1<<<1024, 256, 0, stream>>>(fr, tp, rpart);
    k_reduce2<<<1, 256, 0, stream>>>(rpart, scal);
    k_colband<<<8, 256, 0, stream>>>(fr, band);
    k_winvar<<<441, 256, 0, stream>>>(band, scal, denom);
    k_corr<<<dim3(8, 128), 128, 0, stream>>>(fr, tp, ccpart);
    k_ccred<<<4, 256, 0, stream>>>(ccpart, scal, cc);
    k_peak<<<1, 512, 0, stream>>>(cc, denom, sh, out + (long)H0f * H0f);
    k_trans<<<dim3(96, 96), dim3(32, 8), 0, stream>>>(fr, sh, out);
}